// RetinaNet_35150012350905
// MI455X (gfx1250) — compile-verified
//
#include <hip/hip_runtime.h>
#include <hip/hip_bf16.h>

typedef unsigned int u32;
typedef unsigned long long u64;
typedef float v2f __attribute__((ext_vector_type(2)));
typedef float v8f __attribute__((ext_vector_type(8)));

#define K_TOP 500
#define NPROB 336           // 3 levels * 16 images * 7 fg classes
#define OFF_SCORES 672000   // boxes: 3*16*7*500*4
#define OFF_KEEP   840000
#define OFF_LABELS 1008000

// ---- order-preserving float <-> u32 key map (ascending) ----
__device__ __forceinline__ u32 f2k(float x) {
    u32 u = __float_as_uint(x);
    return u ^ (u32)(((int)u >> 31) | 0x80000000);
}
__device__ __forceinline__ float k2f(u32 k) {
    u32 u = (k & 0x80000000u) ? (k ^ 0x80000000u) : ~k;
    return __uint_as_float(u);
}

// ---- bitonic sorts in LDS (512 elements, 256 threads) ----
__device__ void bitonicAscU32(u32* a, int tid) {
    for (int k = 2; k <= 512; k <<= 1)
        for (int j = k >> 1; j > 0; j >>= 1) {
            __syncthreads();
            for (int i = tid; i < 512; i += 256) {
                int l = i ^ j;
                if (l > i) {
                    bool up = ((i & k) == 0);
                    u32 x = a[i], y = a[l];
                    if ((x > y) == up) { a[i] = y; a[l] = x; }
                }
            }
        }
    __syncthreads();
}
__device__ void bitonicDescU64(u64* a, int tid) {
    for (int k = 2; k <= 512; k <<= 1)
        for (int j = k >> 1; j > 0; j >>= 1) {
            __syncthreads();
            for (int i = tid; i < 512; i += 256) {
                int l = i ^ j;
                if (l > i) {
                    bool up = ((i & k) == 0);
                    u64 x = a[i], y = a[l];
                    if ((x < y) == up) { a[i] = y; a[l] = x; }
                }
            }
        }
    __syncthreads();
}

// ============================================================
// Kernel 1: per-problem exact top-K via 4-pass radix select on
// order-preserving keys of raw logits (sigmoid is monotonic),
// then 512-wide bitonic sort for exact descending order with
// index tie-break. Writes sorted scores + stashes anchor index
// (float-encoded) in the labels region for kernel 2.
// ============================================================
__global__ __launch_bounds__(256)
void topk_kernel(const float* __restrict__ c3, const float* __restrict__ c4,
                 const float* __restrict__ c5, float* __restrict__ out) {
    __shared__ u32 hist[256];
    __shared__ u64 cand[512];
    __shared__ u32 eqb[512];
    __shared__ u32 s_prefix, s_need, s_cgt, s_ceq;

    const int p = blockIdx.x;
    const int l = p / 112, r = p % 112;
    const int b = r / 7, cf = r % 7, c = cf + 1;
    const int HWs[3] = {36864, 9216, 2304};
    const float* clsAll[3] = {c3, c4, c5};
    const float* cls = clsAll[l];
    const int HW = HWs[l];
    const int tid = threadIdx.x;

    if (tid == 0) { s_prefix = 0; s_need = K_TOP; }
    __syncthreads();

    // ---- 4-pass radix select (find K-th largest key) ----
    for (int pass = 0; pass < 4; ++pass) {
        hist[tid] = 0;
        __syncthreads();
        const u32 pref = s_prefix;
        const u32 need = s_need;
        const int shift = 24 - pass * 8;
        for (int a = 0; a < 3; ++a) {
            const float* src = cls + (size_t)(b * 24 + a * 8 + c) * HW;
            for (int hw = tid; hw < HW; hw += 256) {
                __builtin_prefetch(src + hw + 4096, 0, 1);
                u32 key = f2k(src[hw]);
                if (pass == 0 || (key >> (shift + 8)) == pref)
                    atomicAdd(&hist[(key >> shift) & 0xFFu], 1u);
            }
        }
        __syncthreads();
        if (tid == 0) {
            u32 acc = 0; int v = 255;
            for (; v > 0; --v) {
                u32 h = hist[v];
                if (acc + h >= need) break;
                acc += h;
            }
            s_prefix = (pref << 8) | (u32)v;
            s_need = need - acc;
        }
        __syncthreads();
    }
    const u32 T = s_prefix;   // exact K-th key
    const u32 E = s_need;     // how many == T to take (smallest indices)

    // ---- collection pass ----
    if (tid == 0) { s_cgt = 0; s_ceq = 0; }
    for (int i = tid; i < 512; i += 256) { cand[i] = 0ull; eqb[i] = 0xFFFFFFFFu; }
    __syncthreads();
    for (int a = 0; a < 3; ++a) {
        const float* src = cls + (size_t)(b * 24 + a * 8 + c) * HW;
        for (int hw = tid; hw < HW; hw += 256) {
            __builtin_prefetch(src + hw + 4096, 0, 1);
            u32 key = f2k(src[hw]);
            u32 n = (u32)(hw * 3 + a);
            if (key > T) {
                u32 q = atomicAdd(&s_cgt, 1u);
                if (q < K_TOP) cand[q] = ((u64)key << 32) | (u32)(~n);
            } else if (key == T) {
                u32 q = atomicAdd(&s_ceq, 1u);
                if (q < 512) eqb[q] = n;
            }
        }
    }
    __syncthreads();
    const u32 cgt = s_cgt;

    // smallest-index selection among keys == T
    bitonicAscU32(eqb, tid);
    for (u32 e = tid; e < E; e += 256) {
        u32 slot = cgt + e;
        if (slot < K_TOP) cand[slot] = ((u64)T << 32) | (u32)(~eqb[e]);
    }
    __syncthreads();

    // exact descending order (key desc, index asc via ~n)
    bitonicDescU64(cand, tid);

    float* outScores = out + OFF_SCORES;
    float* outIdxStash = out + OFF_LABELS;   // temporary stash, rewritten by kernel 2
    const size_t pb = (size_t)p * K_TOP;
    for (int k = tid; k < K_TOP; k += 256) {
        u64 e = cand[k];
        u32 key = (u32)(e >> 32);
        u32 n = ~(u32)e;
        float logit = k2f(key);
        outScores[pb + k] = 1.0f / (1.0f + expf(-logit));
        outIdxStash[pb + k] = (float)n;      // n < 2^24: exact in fp32
    }
}

// ============================================================
// Kernel 2: decode boxes for selected anchors, build 512x512
// suppression-bit matrix (union term via V_WMMA_F32_16X16X4_F32
// outer sum of areas, ballot -> LDS bitmasks), then wave-wide
// sequential greedy NMS on a register-resident keep mask.
// ============================================================
__global__ __launch_bounds__(512)
void nms_kernel(const float* __restrict__ r3, const float* __restrict__ r4,
                const float* __restrict__ r5, float* __restrict__ out) {
    __shared__ float bx[512][4];
    __shared__ float area[512];
    __shared__ unsigned short adjH[512 * 32];   // 512 rows x 512 bits
    __shared__ u32 validH[32];

    const int p = blockIdx.x;
    const int l = p / 112, rr = p % 112;
    const int b = rr / 7, cf = rr % 7;
    const int Wl[3] = {192, 96, 48};
    const int HWs[3] = {36864, 9216, 2304};
    const float strideL[3] = {8.f, 16.f, 32.f};
    const float sizeL[3] = {16.f, 32.f, 64.f};
    const float* regAll[3] = {r3, r4, r5};
    const float* reg = regAll[l];
    const int W = Wl[l], HW = HWs[l];
    const float stride = strideL[l], asz = sizeL[l];

    const int tid = threadIdx.x;
    const int wv = tid >> 5, lane = tid & 31;
    float* outBoxes = out;
    float* outScores = out + OFF_SCORES;
    float* outKeep = out + OFF_KEEP;
    float* outLabels = out + OFF_LABELS;
    const size_t pb = (size_t)p * K_TOP;

    float sc = 0.f;
    if (tid < K_TOP) {
        u32 n = (u32)outLabels[pb + tid];       // stashed anchor index
        int a = (int)(n % 3u);
        int hw = (int)(n / 3u);
        int wi = hw % W, hi = hw / W;
        const float sq[3] = {0.70710678118f, 1.0f, 1.41421356237f};
        float aw = asz * sq[a], ah = asz / sq[a];
        float cxa = ((float)wi + 0.5f) * stride;
        float cya = ((float)hi + 0.5f) * stride;
        const float* rb = reg + (size_t)(b * 12 + a * 4) * HW + hw;
        float dx = rb[0], dy = rb[(size_t)HW], dw = rb[2 * (size_t)HW], dh = rb[3 * (size_t)HW];
        float cx = dx * aw + cxa, cy = dy * ah + cya;
        float w = expf(dw) * aw, h = expf(dh) * ah;
        float x1 = cx - 0.5f * w, y1 = cy - 0.5f * h;
        float x2 = cx + 0.5f * w, y2 = cy + 0.5f * h;
        bx[tid][0] = x1; bx[tid][1] = y1; bx[tid][2] = x2; bx[tid][3] = y2;
        area[tid] = (x2 - x1) * (y2 - y1);
        outBoxes[pb * 4 + (size_t)tid * 4 + 0] = x1;
        outBoxes[pb * 4 + (size_t)tid * 4 + 1] = y1;
        outBoxes[pb * 4 + (size_t)tid * 4 + 2] = x2;
        outBoxes[pb * 4 + (size_t)tid * 4 + 3] = y2;
        sc = outScores[pb + tid];
        outLabels[pb + tid] = (float)cf;        // final label value
    } else {
        bx[tid][0] = 0.f; bx[tid][1] = 0.f; bx[tid][2] = 0.f; bx[tid][3] = 0.f;
        area[tid] = 0.f;
    }
    bool valid = (tid < K_TOP) && (sc > 0.05f);
    u32 bal = (u32)__ballot(valid);
    if (lane == 0) { validH[wv * 2] = bal & 0xffffu; validH[wv * 2 + 1] = bal >> 16; }
    __syncthreads();

    // ---- adjacency: 32x32 grid of 16x16 tiles, union base via WMMA ----
    for (int t = wv; t < 1024; t += 16) {
        const int ti = t >> 5, tj = t & 31;
        // A (16x4): col0 = area_i, col1 = 1; K=2,3 (lanes 16..31) zero.
        // B (4x16): row0 = 1,    row1 = area_j; K=2,3 (lanes 16..31) zero.
        float ai = (lane < 16) ? area[ti * 16 + (lane & 15)] : 0.f;
        float aj = (lane < 16) ? area[tj * 16 + (lane & 15)] : 0.f;
        v2f A = {ai, (lane < 16) ? 1.f : 0.f};
        v2f Bv = {(lane < 16) ? 1.f : 0.f, aj};
        v8f Cv = {0.f, 0.f, 0.f, 0.f, 0.f, 0.f, 0.f, 0.f};
        // D[m][n] = area_i[m] + area_j[n]
        v8f S = __builtin_amdgcn_wmma_f32_16x16x4_f32(
            false, A, false, Bv, (short)0, Cv, false, false);

        const int jj = tj * 16 + (lane & 15);
        const float jx1 = bx[jj][0], jy1 = bx[jj][1], jx2 = bx[jj][2], jy2 = bx[jj][3];
        const int ibase = ti * 16 + ((lane >> 4) << 3);   // rows r (lanes<16), r+8 (lanes>=16)
#pragma unroll
        for (int r2 = 0; r2 < 8; ++r2) {
            const int ii = ibase + r2;
            float iw = fminf(bx[ii][2], jx2) - fmaxf(bx[ii][0], jx1);
            float ih = fminf(bx[ii][3], jy2) - fmaxf(bx[ii][1], jy1);
            float inter = fmaxf(iw, 0.f) * fmaxf(ih, 0.f);
            float uni = S[r2] - inter;
            bool sup = inter > 0.5f * fmaxf(uni, 1e-9f);
            u32 bb = (u32)__ballot(sup);
            if (lane == 0) {
                adjH[(ti * 16 + r2) * 32 + tj] = (unsigned short)(bb & 0xffffu);
                adjH[(ti * 16 + r2 + 8) * 32 + tj] = (unsigned short)(bb >> 16);
            }
        }
    }
    __syncthreads();

    // ---- sequential greedy NMS: wave 0, 512-bit keep mask in registers ----
    if (wv == 0) {
        u32 keep = validH[lane] & 0xffffu;     // lane owns bits [16*lane, 16*lane+16)
        for (int i = 0; i < K_TOP; ++i) {
            const int word = i >> 4;
            u32 kw = (u32)__shfl((int)keep, word, 32);
            if ((kw >> (i & 15)) & 1u) {
                u32 m = (u32)adjH[i * 32 + lane];
                u32 mask;
                if (lane < word) mask = 0u;
                else if (lane > word) mask = 0xffffu;
                else mask = (0xffffu << ((i & 15) + 1)) & 0xffffu;   // only j > i
                keep &= ~(m & mask);
            }
        }
        for (int bit = 0; bit < 16; ++bit) {
            int k = lane * 16 + bit;
            if (k < K_TOP) outKeep[pb + k] = ((keep >> bit) & 1u) ? 1.0f : 0.0f;
        }
    }
}

extern "C" void kernel_launch(void* const* d_in, const int* in_sizes, int n_in,
                              void* d_out, int out_size, void* d_ws, size_t ws_size,
                              hipStream_t stream) {
    // setup_inputs() dict order: cls_p3, reg_p3, cls_p4, reg_p4, cls_p5, reg_p5
    const float* c3 = (const float*)d_in[0];
    const float* r3 = (const float*)d_in[1];
    const float* c4 = (const float*)d_in[2];
    const float* r4 = (const float*)d_in[3];
    const float* c5 = (const float*)d_in[4];
    const float* r5 = (const float*)d_in[5];
    float* out = (float*)d_out;

    topk_kernel<<<NPROB, 256, 0, stream>>>(c3, c4, c5, out);
    nms_kernel<<<NPROB, 512, 0, stream>>>(r3, r4, r5, out);
    (void)in_sizes; (void)n_in; (void)out_size; (void)d_ws; (void)ws_size;
}